// Model_48868137894217
// MI455X (gfx1250) — compile-verified
//
#include <hip/hip_runtime.h>
#include <math.h>

// D = 128 feature dim; wave32; 128-thread blocks (4 waves).
#define D 128
#define DPAD 132   // LDS row stride for 128-wide tiles (pad vs bank conflicts)
#define GPAD 260   // LDS row stride for r/z sum (256 cols)

typedef float v2f __attribute__((ext_vector_type(2)));
typedef float v8f __attribute__((ext_vector_type(8)));

static __device__ __forceinline__ v8f zero8() {
  v8f z = {0.f, 0.f, 0.f, 0.f, 0.f, 0.f, 0.f, 0.f};
  return z;
}

// CODES = [3,2,5,1,4]  (AND, NOT, XOR, MAJ, OR)
static __device__ __forceinline__ int t_of_code(int g) {
  switch (g) {
    case 3: return 0;
    case 2: return 1;
    case 5: return 2;
    case 1: return 3;
    default: return 4;  // code 4
  }
}

// 16x16 fp32 tile GEMM: C += A(16xK, LDS, stride lda) * B(KxNb, global, stride ldb),
// column tile at cb. Uses V_WMMA_F32_16X16X4_F32, K stepped by 4.
static __device__ __forceinline__ v8f gemm_tile_f32(const float* A, int lda,
                                                    const float* B, int ldb,
                                                    int K, int cb, int lane,
                                                    v8f c) {
  const int r  = lane & 15;
  const int ko = (lane >> 4) << 1;  // 0 or 2
  const float* arow = A + r * lda + ko;
  const float* bcol = B + (size_t)ko * ldb + cb + (lane & 15);
#pragma unroll 4
  for (int kb = 0; kb < K; kb += 4) {
    v2f a;
    a.x = arow[kb];
    a.y = arow[kb + 1];
    v2f b;
    b.x = bcol[(size_t)kb * ldb];
    b.y = bcol[(size_t)(kb + 1) * ldb];
    c = __builtin_amdgcn_wmma_f32_16x16x4_f32(false, a, false, b, (short)0, c,
                                              false, false);
  }
  return c;
}

// ---------------- bucketing / setup kernels ----------------

__global__ void k_zero_counts(int* ecount, int* ncount) {
  int i = threadIdx.x;
  if (i < 20) { ecount[i] = 0; ncount[i] = 0; }
}

__global__ void k_count(const int* __restrict__ edst,
                        const int* __restrict__ gate,
                        const int* __restrict__ flevel, int E, int N,
                        int* ecount, int* ncount) {
  int i = blockIdx.x * blockDim.x + threadIdx.x;
  if (i < E) {
    int d = edst[i];
    int lv = flevel[d], g = gate[d];
    if (lv >= 1 && g >= 1 && g <= 5)
      atomicAdd(&ecount[(lv - 1) * 5 + t_of_code(g)], 1);
  }
  if (i < N) {
    int lv = flevel[i], g = gate[i];
    if (lv >= 1 && g >= 1 && g <= 5)
      atomicAdd(&ncount[(lv - 1) * 5 + t_of_code(g)], 1);
  }
}

__global__ void k_scan(const int* ecount, int* eoff, int* ecur,
                       const int* ncount, int* noff, int* ncur) {
  if (threadIdx.x == 0 && blockIdx.x == 0) {
    int a = 0;
    for (int i = 0; i < 20; i++) { eoff[i] = a; ecur[i] = a; a += ecount[i]; }
    eoff[20] = a;
    a = 0;
    for (int i = 0; i < 20; i++) { noff[i] = a; ncur[i] = a; a += ncount[i]; }
    noff[20] = a;
  }
}

__global__ void k_scatter(const int* __restrict__ edst,
                          const int* __restrict__ gate,
                          const int* __restrict__ flevel, int E, int N,
                          int* ecur, int* bucketEdges, int* ncur,
                          int* nodeList) {
  int i = blockIdx.x * blockDim.x + threadIdx.x;
  if (i < E) {
    int d = edst[i];
    int lv = flevel[d], g = gate[d];
    if (lv >= 1 && g >= 1 && g <= 5) {
      int slot = atomicAdd(&ecur[(lv - 1) * 5 + t_of_code(g)], 1);
      bucketEdges[slot] = i;
    }
  }
  if (i < N) {
    int lv = flevel[i], g = gate[i];
    if (lv >= 1 && g >= 1 && g <= 5) {
      int slot = atomicAdd(&ncur[(lv - 1) * 5 + t_of_code(g)], 1);
      nodeList[slot] = i;
    }
  }
}

__global__ void k_init_state(const float* __restrict__ hs_init,
                             const int* __restrict__ gate,
                             float* __restrict__ hs, float* __restrict__ hf,
                             int N) {
  int i = blockIdx.x * blockDim.x + threadIdx.x;
  if (i < N * D) {
    int n = i >> 7;
    hs[i] = (gate[n] == 0) ? hs_init[i] : 0.f;
    hf[i] = 0.f;
  }
}

__global__ void k_zero_agg(float* aggS, float* aggF, int total) {
  int i = blockIdx.x * blockDim.x + threadIdx.x;
  if (i < total) { aggS[i] = 0.f; aggF[i] = 0.f; }
}

// ---------------- per-level message kernel ----------------
// grid = (ceil(E/16), 5); block = 128 (4 wave32s). Each block: 16 edges of
// bucket (level, t). Computes strc + func 2-layer MLP messages with WMMA and
// atomic-scatters rows into aggS/aggF[dst].

__global__ void __launch_bounds__(128) k_messages(
    const float* __restrict__ hs, const float* __restrict__ hf,
    const float* __restrict__ Ws1, const float* __restrict__ bs1,
    const float* __restrict__ Ws2, const float* __restrict__ bs2,
    const float* __restrict__ Wf1, const float* __restrict__ bf1,
    const float* __restrict__ Wf2, const float* __restrict__ bf2,
    const float* __restrict__ Wnf1, const float* __restrict__ bnf1,
    const float* __restrict__ Wnf2, const float* __restrict__ bnf2,
    const int* __restrict__ esrc, const int* __restrict__ edst,
    const int* __restrict__ eoff, const int* __restrict__ bucketEdges,
    float* __restrict__ aggS, float* __restrict__ aggF, int level) {
  const int t = blockIdx.y;
  const int b = (level - 1) * 5 + t;
  const int begin = eoff[b];
  const int nb = eoff[b + 1] - begin;
  if ((int)blockIdx.x * 16 >= nb) return;

  __shared__ float Xs[16][DPAD];
  __shared__ float Xf[16][DPAD];
  __shared__ float H1[16][DPAD];
  __shared__ int sSrc[16];
  __shared__ int sDst[16];

  const int tid = threadIdx.x;
  const int lane = tid & 31;
  const int wave = tid >> 5;

  if (tid < 16) {
    int idx = blockIdx.x * 16 + tid;
    int e = (idx < nb) ? bucketEdges[begin + idx] : -1;
    sSrc[tid] = (e >= 0) ? esrc[e] : -1;
    sDst[tid] = (e >= 0) ? edst[e] : -1;
  }
  __syncthreads();

  // gather hs[src], hf[src] rows (padding rows -> 0)
  {
    const int row = tid >> 3;
    const int c0 = (tid & 7) * 16;
    const int s = sSrc[row];
#pragma unroll
    for (int c = 0; c < 16; c += 4) {
      float4 vs = make_float4(0.f, 0.f, 0.f, 0.f);
      float4 vf = vs;
      if (s >= 0) {
        vs = *(const float4*)&hs[(size_t)s * D + c0 + c];
        vf = *(const float4*)&hf[(size_t)s * D + c0 + c];
      }
      *(float4*)&Xs[row][c0 + c] = vs;
      *(float4*)&Xf[row][c0 + c] = vf;
    }
  }
  __syncthreads();

  const int cb0 = (wave * 2) * 16;
  const int cb1 = cb0 + 16;
  const int colA = lane & 15;
  const int mo = (lane >> 4) * 8;

  // -------- structural message: relu(Xs@Ws1+bs1)@Ws2+bs2 -> aggS[dst] -----
  {
    const float* W1 = Ws1 + (size_t)t * D * D;
    const float* W2 = Ws2 + (size_t)t * D * D;
    const float* b1 = bs1 + t * D;
    const float* b2 = bs2 + t * D;
    v8f c0 = zero8(), c1 = zero8();
    c0 = gemm_tile_f32(&Xs[0][0], DPAD, W1, D, D, cb0, lane, c0);
    c1 = gemm_tile_f32(&Xs[0][0], DPAD, W1, D, D, cb1, lane, c1);
#pragma unroll
    for (int i = 0; i < 8; i++) {
      float v0 = c0[i] + b1[cb0 + colA];
      float v1 = c1[i] + b1[cb1 + colA];
      H1[i + mo][cb0 + colA] = v0 > 0.f ? v0 : 0.f;
      H1[i + mo][cb1 + colA] = v1 > 0.f ? v1 : 0.f;
    }
    __syncthreads();
    c0 = zero8(); c1 = zero8();
    c0 = gemm_tile_f32(&H1[0][0], DPAD, W2, D, D, cb0, lane, c0);
    c1 = gemm_tile_f32(&H1[0][0], DPAD, W2, D, D, cb1, lane, c1);
#pragma unroll
    for (int i = 0; i < 8; i++) {
      int dn = sDst[i + mo];
      if (dn >= 0) {
        atomicAdd(&aggS[(size_t)dn * D + cb0 + colA], c0[i] + b2[cb0 + colA]);
        atomicAdd(&aggS[(size_t)dn * D + cb1 + colA], c1[i] + b2[cb1 + colA]);
      }
    }
    __syncthreads();
  }

  // -------- functional message -> aggF[dst] --------
  if (t == 1) {  // NOT: relu(Xf@Wnf1+bnf1)@Wnf2+bnf2
    v8f c0 = zero8(), c1 = zero8();
    c0 = gemm_tile_f32(&Xf[0][0], DPAD, Wnf1, D, D, cb0, lane, c0);
    c1 = gemm_tile_f32(&Xf[0][0], DPAD, Wnf1, D, D, cb1, lane, c1);
#pragma unroll
    for (int i = 0; i < 8; i++) {
      float v0 = c0[i] + bnf1[cb0 + colA];
      float v1 = c1[i] + bnf1[cb1 + colA];
      H1[i + mo][cb0 + colA] = v0 > 0.f ? v0 : 0.f;
      H1[i + mo][cb1 + colA] = v1 > 0.f ? v1 : 0.f;
    }
    __syncthreads();
    c0 = zero8(); c1 = zero8();
    c0 = gemm_tile_f32(&H1[0][0], DPAD, Wnf2, D, D, cb0, lane, c0);
    c1 = gemm_tile_f32(&H1[0][0], DPAD, Wnf2, D, D, cb1, lane, c1);
#pragma unroll
    for (int i = 0; i < 8; i++) {
      int dn = sDst[i + mo];
      if (dn >= 0) {
        atomicAdd(&aggF[(size_t)dn * D + cb0 + colA], c0[i] + bnf2[cb0 + colA]);
        atomicAdd(&aggF[(size_t)dn * D + cb1 + colA], c1[i] + bnf2[cb1 + colA]);
      }
    }
  } else {  // relu([Xs,Xf]@Wf1+bf1)@Wf2+bf2, K=256 split across Xs/Xf
    const int fi = (t == 0) ? 0 : (t == 2) ? 1 : (t == 3) ? 2 : 3;
    const float* W1 = Wf1 + (size_t)fi * 2 * D * D;  // rows 0..127: hs part
    const float* W2 = Wf2 + (size_t)fi * D * D;
    const float* b1 = bf1 + fi * D;
    const float* b2 = bf2 + fi * D;
    v8f c0 = zero8(), c1 = zero8();
    c0 = gemm_tile_f32(&Xs[0][0], DPAD, W1, D, D, cb0, lane, c0);
    c1 = gemm_tile_f32(&Xs[0][0], DPAD, W1, D, D, cb1, lane, c1);
    c0 = gemm_tile_f32(&Xf[0][0], DPAD, W1 + (size_t)D * D, D, D, cb0, lane, c0);
    c1 = gemm_tile_f32(&Xf[0][0], DPAD, W1 + (size_t)D * D, D, D, cb1, lane, c1);
#pragma unroll
    for (int i = 0; i < 8; i++) {
      float v0 = c0[i] + b1[cb0 + colA];
      float v1 = c1[i] + b1[cb1 + colA];
      H1[i + mo][cb0 + colA] = v0 > 0.f ? v0 : 0.f;
      H1[i + mo][cb1 + colA] = v1 > 0.f ? v1 : 0.f;
    }
    __syncthreads();
    c0 = zero8(); c1 = zero8();
    c0 = gemm_tile_f32(&H1[0][0], DPAD, W2, D, D, cb0, lane, c0);
    c1 = gemm_tile_f32(&H1[0][0], DPAD, W2, D, D, cb1, lane, c1);
#pragma unroll
    for (int i = 0; i < 8; i++) {
      int dn = sDst[i + mo];
      if (dn >= 0) {
        atomicAdd(&aggF[(size_t)dn * D + cb0 + colA], c0[i] + b2[cb0 + colA]);
        atomicAdd(&aggF[(size_t)dn * D + cb1 + colA], c1[i] + b2[cb1 + colA]);
      }
    }
  }
}

// ---------------- per-level GRU kernel ----------------
// grid = (ceil(N/16), 5). Each block: 16 active nodes of bucket (level, t).
// pass 0: hs <- GRU(aggS, hs; Gs[t]); pass 1: hf <- GRU(aggF, hf; Gf[t]).

__global__ void __launch_bounds__(128) k_gru(
    float* __restrict__ hs, float* __restrict__ hf,
    const float* __restrict__ aggS, const float* __restrict__ aggF,
    const float* __restrict__ Gs_wih, const float* __restrict__ Gs_whh,
    const float* __restrict__ Gs_bih, const float* __restrict__ Gs_bhh,
    const float* __restrict__ Gf_wih, const float* __restrict__ Gf_whh,
    const float* __restrict__ Gf_bih, const float* __restrict__ Gf_bhh,
    const int* __restrict__ noff, const int* __restrict__ nodeList,
    int level) {
  const int t = blockIdx.y;
  const int b = (level - 1) * 5 + t;
  const int begin = noff[b];
  const int nb = noff[b + 1] - begin;
  if ((int)blockIdx.x * 16 >= nb) return;

  __shared__ float X[16][DPAD];    // gru input (agg)
  __shared__ float H[16][DPAD];    // hidden state
  __shared__ float SRZ[16][GPAD];  // gi+gh for r (cols 0..127) and z (128..255)
  __shared__ float INN[16][DPAD];  // gi "n" part
  __shared__ float HNN[16][DPAD];  // gh "n" part
  __shared__ int sN[16];

  const int tid = threadIdx.x;
  const int lane = tid & 31;
  const int wave = tid >> 5;

  if (tid < 16) {
    int idx = blockIdx.x * 16 + tid;
    sN[tid] = (idx < nb) ? nodeList[begin + idx] : -1;
  }
  __syncthreads();

  for (int pass = 0; pass < 2; ++pass) {
    const float* agg = pass ? aggF : aggS;
    float* state = pass ? hf : hs;
    const float* wih = (pass ? Gf_wih : Gs_wih) + (size_t)t * D * 3 * D;
    const float* whh = (pass ? Gf_whh : Gs_whh) + (size_t)t * D * 3 * D;
    const float* bih = (pass ? Gf_bih : Gs_bih) + t * 3 * D;
    const float* bhh = (pass ? Gf_bhh : Gs_bhh) + t * 3 * D;

    {
      const int row = tid >> 3;
      const int c0 = (tid & 7) * 16;
      const int n = sN[row];
#pragma unroll
      for (int c = 0; c < 16; c += 4) {
        float4 vx = make_float4(0.f, 0.f, 0.f, 0.f);
        float4 vh = vx;
        if (n >= 0) {
          vx = *(const float4*)&agg[(size_t)n * D + c0 + c];
          vh = *(const float4*)&state[(size_t)n * D + c0 + c];
        }
        *(float4*)&X[row][c0 + c] = vx;
        *(float4*)&H[row][c0 + c] = vh;
      }
    }
    __syncthreads();

    const int colA = lane & 15;
    const int mo = (lane >> 4) * 8;
    for (int ct = wave; ct < 24; ct += 4) {  // 24 col tiles of 3D=384 output
      const int cb = ct * 16;
      v8f ci = zero8(), ch = zero8();
      ci = gemm_tile_f32(&X[0][0], DPAD, wih, 3 * D, D, cb, lane, ci);
      ch = gemm_tile_f32(&H[0][0], DPAD, whh, 3 * D, D, cb, lane, ch);
      const int cg = cb + colA;
      if (ct < 16) {
#pragma unroll
        for (int i = 0; i < 8; i++)
          SRZ[i + mo][cg] = (ci[i] + bih[cg]) + (ch[i] + bhh[cg]);
      } else {
#pragma unroll
        for (int i = 0; i < 8; i++) {
          INN[i + mo][cg - 256] = ci[i] + bih[cg];
          HNN[i + mo][cg - 256] = ch[i] + bhh[cg];
        }
      }
    }
    __syncthreads();

    // elementwise GRU gates + writeback
    {
      const int row = tid >> 3;
      const int c0 = (tid & 7) * 16;
      const int n = sN[row];
      if (n >= 0) {
#pragma unroll 4
        for (int c = 0; c < 16; c++) {
          int d = c0 + c;
          float r = 1.f / (1.f + expf(-SRZ[row][d]));
          float z = 1.f / (1.f + expf(-SRZ[row][128 + d]));
          float nn = tanhf(INN[row][d] + r * HNN[row][d]);
          state[(size_t)n * D + d] = (1.f - z) * nn + z * H[row][d];
        }
      }
    }
    __syncthreads();
  }
}

// ---------------- host ----------------

extern "C" void kernel_launch(void* const* d_in, const int* in_sizes, int n_in,
                              void* d_out, int out_size, void* d_ws,
                              size_t ws_size, hipStream_t stream) {
  const float* hs_init = (const float*)d_in[0];
  const float* Ws1 = (const float*)d_in[1];
  const float* bs1 = (const float*)d_in[2];
  const float* Ws2 = (const float*)d_in[3];
  const float* bs2 = (const float*)d_in[4];
  const float* Wf1 = (const float*)d_in[5];
  const float* bf1 = (const float*)d_in[6];
  const float* Wf2 = (const float*)d_in[7];
  const float* bf2 = (const float*)d_in[8];
  const float* Wnf1 = (const float*)d_in[9];
  const float* bnf1 = (const float*)d_in[10];
  const float* Wnf2 = (const float*)d_in[11];
  const float* bnf2 = (const float*)d_in[12];
  const float* Gs_wih = (const float*)d_in[13];
  const float* Gs_whh = (const float*)d_in[14];
  const float* Gs_bih = (const float*)d_in[15];
  const float* Gs_bhh = (const float*)d_in[16];
  const float* Gf_wih = (const float*)d_in[17];
  const float* Gf_whh = (const float*)d_in[18];
  const float* Gf_bih = (const float*)d_in[19];
  const float* Gf_bhh = (const float*)d_in[20];
  const int* edge_index = (const int*)d_in[21];
  const int* gate = (const int*)d_in[22];
  const int* flevel = (const int*)d_in[23];

  const int N = in_sizes[0] / D;
  const int E = in_sizes[21] / 2;
  const int L = 5;  // num_levels (fixed by setup_inputs)
  const int* esrc = edge_index;
  const int* edst = edge_index + E;

  float* hs = (float*)d_out;
  float* hf = hs + (size_t)N * D;

  // workspace carve-out
  char* w = (char*)d_ws;
  size_t off = 0;
  auto carve = [&](size_t bytes) -> void* {
    void* p = w + off;
    off = (off + bytes + 255) & ~(size_t)255;
    return p;
  };
  int* ecount = (int*)carve(20 * sizeof(int));
  int* eoff = (int*)carve(21 * sizeof(int));
  int* ecur = (int*)carve(20 * sizeof(int));
  int* ncount = (int*)carve(20 * sizeof(int));
  int* noff = (int*)carve(21 * sizeof(int));
  int* ncur = (int*)carve(20 * sizeof(int));
  int* bucketEdges = (int*)carve((size_t)E * sizeof(int));
  int* nodeList = (int*)carve((size_t)N * sizeof(int));
  float* aggS = (float*)carve((size_t)N * D * sizeof(float));
  float* aggF = (float*)carve((size_t)N * D * sizeof(float));
  (void)ws_size; (void)n_in; (void)out_size;

  // setup: bucket edges/nodes by (level, gate-type), init states
  k_zero_counts<<<1, 32, 0, stream>>>(ecount, ncount);
  {
    int M = E > N ? E : N;
    int blk = (M + 255) / 256;
    k_count<<<blk, 256, 0, stream>>>(edst, gate, flevel, E, N, ecount, ncount);
    k_scan<<<1, 32, 0, stream>>>(ecount, eoff, ecur, ncount, noff, ncur);
    k_scatter<<<blk, 256, 0, stream>>>(edst, gate, flevel, E, N, ecur,
                                       bucketEdges, ncur, nodeList);
  }
  {
    int total = N * D;
    k_init_state<<<(total + 255) / 256, 256, 0, stream>>>(hs_init, gate, hs,
                                                          hf, N);
  }

  dim3 mgrid((E + 15) / 16, 5);
  dim3 ggrid((N + 15) / 16, 5);
  int aggTotal = N * D;
  for (int level = 1; level < L; ++level) {
    k_zero_agg<<<(aggTotal + 255) / 256, 256, 0, stream>>>(aggS, aggF,
                                                           aggTotal);
    k_messages<<<mgrid, 128, 0, stream>>>(
        hs, hf, Ws1, bs1, Ws2, bs2, Wf1, bf1, Wf2, bf2, Wnf1, bnf1, Wnf2,
        bnf2, esrc, edst, eoff, bucketEdges, aggS, aggF, level);
    k_gru<<<ggrid, 128, 0, stream>>>(hs, hf, aggS, aggF, Gs_wih, Gs_whh,
                                     Gs_bih, Gs_bhh, Gf_wih, Gf_whh, Gf_bih,
                                     Gf_bhh, noff, nodeList, level);
  }
}